// GATConvBlock_26714696581259
// MI455X (gfx1250) — compile-verified
//
#include <hip/hip_runtime.h>
#include <hip/hip_bf16.h>

#define N_NODES 50000
#define N_EDGES 1600000
#define E_TOT   (N_EDGES + N_NODES)
#define IN_CH   128
#define OUT_CH  128
#define HEADS   4
#define HEAD_DIM 32
#define NEG_SLOPE 0.2f
#define WT_PITCH 132   // transposed-W LDS row pitch (mod 64 == 4 -> conflict-free b64 loads)

typedef float v2f __attribute__((ext_vector_type(2)));
typedef float v8f __attribute__((ext_vector_type(8)));

// ---------------------------------------------------------------------------
// Kernel 1: h = x @ W via V_WMMA_F32_16X16X4_F32 (exact fp32 WMMA).
// Block = 256 threads (8 waves). W is staged TRANSPOSED in LDS (pitch 132)
// so every B fragment {W[k][col], W[k+1][col]} is one aligned ds_load_b64
// landing directly in an even VGPR pair -> no shuffle movs before v_wmma.
// Each wave owns a 16-row slab of x and all 8 column tiles (8 accumulators),
// looping K in steps of 4 (32 WMMA issues per tile -> 256 wmma per wave).
// A-frag layout (ISA 7.12.2, 32-bit A 16x4): lane<16 holds (M=lane, K=0,1),
// lane>=16 holds (M=lane-16, K=2,3)  -> contiguous float2 load per lane.
// C/D layout: VGPR j -> M = j + 8*(lane>>4), N = lane&15.
// ---------------------------------------------------------------------------
__global__ __launch_bounds__(256)
void gat_gemm_wmma(const float* __restrict__ x, const float* __restrict__ W,
                   float* __restrict__ h) {
  __shared__ float sWt[OUT_CH * WT_PITCH];  // 66 KB, W transposed: sWt[col*132+k]
  const int tid = threadIdx.x;

  // cooperative transposed stage of W (coalesced global reads)
  {
#pragma unroll
    for (int i = 0; i < (IN_CH * OUT_CH) / 256; ++i) {
      const int idx = tid + i * 256;
      const int k = idx >> 7;          // row of W
      const int col = idx & 127;       // col of W
      sWt[col * WT_PITCH + k] = W[idx];
    }
  }
  __syncthreads();

  const int wave = tid >> 5;
  const int lane = tid & 31;
  const int m0 = (blockIdx.x * 8 + wave) * 16;   // 50000 % 16 == 0, no tail rows
  if (m0 >= N_NODES) return;                     // wave-uniform: EXEC stays all-1s

  const int r  = lane & 15;          // row within tile (A) / col within tile (B,C)
  const int ko = (lane >> 4) * 2;    // K offset 0 or 2

  v8f acc[8] = {};                   // 8 tiles x 8 VGPRs = 64 accumulator VGPRs

  const float* xrow = x + (size_t)(m0 + r) * IN_CH + ko;

  for (int kk = 0; kk < IN_CH / 4; ++kk) {
    v2f a;
    a.x = xrow[kk * 4 + 0];          // (M=r, K=4kk+ko)
    a.y = xrow[kk * 4 + 1];          // (M=r, K=4kk+ko+1)
#pragma unroll
    for (int t = 0; t < 8; ++t) {
      const int col = t * 16 + r;
      // single aligned ds_load_b64: {W[4kk+ko][col], W[4kk+ko+1][col]}
      const v2f b = *(const v2f*)&sWt[col * WT_PITCH + kk * 4 + ko];
      acc[t] = __builtin_amdgcn_wmma_f32_16x16x4_f32(
          false, a, false, b, (short)0, acc[t], false, false);
    }
  }

  const int mhi = (lane >> 4) * 8;
#pragma unroll
  for (int t = 0; t < 8; ++t) {
#pragma unroll
    for (int j = 0; j < 8; ++j) {
      const int m = j + mhi;
      h[(size_t)(m0 + m) * OUT_CH + t * 16 + r] = acc[t][j];
    }
  }
}

// ---------------------------------------------------------------------------
// Kernel 2: per-(node,head) attention logits a_src/a_dst; also re-initializes
// amax (-inf), denom (0) and the scatter accumulator each call (replay-safe).
// ---------------------------------------------------------------------------
__global__ __launch_bounds__(256)
void gat_logits_init(const float* __restrict__ h,
                     const float* __restrict__ att_src,
                     const float* __restrict__ att_dst,
                     float* __restrict__ a_src, float* __restrict__ a_dst,
                     float* __restrict__ amax, float* __restrict__ denom,
                     float* __restrict__ acc) {
  const int i = blockIdx.x * blockDim.x + threadIdx.x;   // n*HEADS + head
  if (i >= N_NODES * HEADS) return;
  const int head = i & (HEADS - 1);
  const int n = i >> 2;
  const float* hv = h + (size_t)n * OUT_CH + head * HEAD_DIM;
  const float* as = att_src + head * HEAD_DIM;
  const float* ad = att_dst + head * HEAD_DIM;
  float s0 = 0.f, s1 = 0.f;
#pragma unroll 8
  for (int c = 0; c < HEAD_DIM; ++c) {
    const float v = hv[c];
    s0 += v * as[c];
    s1 += v * ad[c];
  }
  a_src[i] = s0;
  a_dst[i] = s1;
  amax[i] = -__builtin_inff();
  denom[i] = 0.f;
  float* av = acc + (size_t)n * OUT_CH + head * HEAD_DIM;
#pragma unroll 8
  for (int c = 0; c < HEAD_DIM; ++c) av[c] = 0.f;
}

// float atomic max via int/uint ordering trick (valid with -inf init)
__device__ __forceinline__ void atomicMaxF(float* addr, float val) {
  if (val >= 0.f)
    atomicMax((int*)addr, __float_as_int(val));
  else
    atomicMin((unsigned int*)addr, __float_as_uint(val));
}

__device__ __forceinline__ void edge_src_dst(const int* __restrict__ ei, int e,
                                             int& s, int& d) {
  if (e < N_EDGES) { s = ei[e]; d = ei[N_EDGES + e]; }
  else             { s = d = e - N_EDGES; }          // self loop
}

// ---------------------------------------------------------------------------
// Kernel 3: segment max of leaky-relu logits per (dst, head).
// ---------------------------------------------------------------------------
__global__ __launch_bounds__(256)
void gat_edge_max(const int* __restrict__ ei,
                  const float* __restrict__ a_src,
                  const float* __restrict__ a_dst,
                  float* __restrict__ amax) {
  const int e = blockIdx.x * blockDim.x + threadIdx.x;
  if (e >= E_TOT) return;
  int s, d;
  edge_src_dst(ei, e, s, d);
#pragma unroll
  for (int hd = 0; hd < HEADS; ++hd) {
    float al = a_src[s * HEADS + hd] + a_dst[d * HEADS + hd];
    al = al > 0.f ? al : al * NEG_SLOPE;
    atomicMaxF(&amax[d * HEADS + hd], al);
  }
}

// ---------------------------------------------------------------------------
// Kernel 4: segment softmax denominator (recompute logit, no alpha buffer).
// ---------------------------------------------------------------------------
__global__ __launch_bounds__(256)
void gat_edge_denom(const int* __restrict__ ei,
                    const float* __restrict__ a_src,
                    const float* __restrict__ a_dst,
                    const float* __restrict__ amax,
                    float* __restrict__ denom) {
  const int e = blockIdx.x * blockDim.x + threadIdx.x;
  if (e >= E_TOT) return;
  int s, d;
  edge_src_dst(ei, e, s, d);
#pragma unroll
  for (int hd = 0; hd < HEADS; ++hd) {
    float al = a_src[s * HEADS + hd] + a_dst[d * HEADS + hd];
    al = al > 0.f ? al : al * NEG_SLOPE;
    atomicAdd(&denom[d * HEADS + hd], __expf(al - amax[d * HEADS + hd]));
  }
}

// ---------------------------------------------------------------------------
// Kernel 5: weighted scatter-aggregate. One thread per (edge, channel);
// 32 consecutive lanes share one (edge, head) so the scalar gathers broadcast
// out of L1/L2. acc (25.6 MB) is L2-resident (192 MB L2) -> L2-local atomics.
// ---------------------------------------------------------------------------
__global__ __launch_bounds__(256)
void gat_edge_agg(const int* __restrict__ ei,
                  const float* __restrict__ a_src,
                  const float* __restrict__ a_dst,
                  const float* __restrict__ amax,
                  const float* __restrict__ denom,
                  const float* __restrict__ h,
                  float* __restrict__ acc) {
  const long long idx = (long long)blockIdx.x * blockDim.x + threadIdx.x;
  if (idx >= (long long)E_TOT * OUT_CH) return;
  const int c = (int)(idx & (OUT_CH - 1));
  const int e = (int)(idx >> 7);
  int s, d;
  edge_src_dst(ei, e, s, d);
  const int hd = c >> 5;
  float al = a_src[s * HEADS + hd] + a_dst[d * HEADS + hd];
  al = al > 0.f ? al : al * NEG_SLOPE;
  const float w = __expf(al - amax[d * HEADS + hd]) /
                  (denom[d * HEADS + hd] + 1e-16f);
  atomicAdd(&acc[(size_t)d * OUT_CH + c], w * h[(size_t)s * OUT_CH + c]);
}

// ---------------------------------------------------------------------------
// Kernel 6: out = elu(acc + bias)
// ---------------------------------------------------------------------------
__global__ __launch_bounds__(256)
void gat_finalize(const float* __restrict__ acc, const float* __restrict__ bias,
                  float* __restrict__ out) {
  const int i = blockIdx.x * blockDim.x + threadIdx.x;
  if (i >= N_NODES * OUT_CH) return;
  const float v = acc[i] + bias[i & (OUT_CH - 1)];
  out[i] = v > 0.f ? v : (__expf(v) - 1.f);
}

extern "C" void kernel_launch(void* const* d_in, const int* in_sizes, int n_in,
                              void* d_out, int out_size, void* d_ws, size_t ws_size,
                              hipStream_t stream) {
  const float* x       = (const float*)d_in[0];
  const int*   ei      = (const int*)d_in[1];   // edge_index [2, E] flattened
  const float* W       = (const float*)d_in[2];
  const float* att_src = (const float*)d_in[3];
  const float* att_dst = (const float*)d_in[4];
  const float* bias    = (const float*)d_in[5];
  float* out = (float*)d_out;

  // workspace layout (floats)
  float* ws    = (float*)d_ws;
  float* h     = ws;                                   // N*128
  float* a_src = h + (size_t)N_NODES * OUT_CH;         // N*4
  float* a_dst = a_src + (size_t)N_NODES * HEADS;      // N*4
  float* amax  = a_dst + (size_t)N_NODES * HEADS;      // N*4
  float* denom = amax + (size_t)N_NODES * HEADS;       // N*4
  float* acc   = denom + (size_t)N_NODES * HEADS;      // N*128

  // 1) projection GEMM (WMMA fp32): 3125 row-tiles, 8 per block
  {
    const int mtiles = N_NODES / 16;            // 3125
    const int blocks = (mtiles + 7) / 8;        // 391
    gat_gemm_wmma<<<blocks, 256, 0, stream>>>(x, W, h);
  }
  // 2) logits + init
  {
    const int n = N_NODES * HEADS;
    gat_logits_init<<<(n + 255) / 256, 256, 0, stream>>>(
        h, att_src, att_dst, a_src, a_dst, amax, denom, acc);
  }
  // 3) segment max
  gat_edge_max<<<(E_TOT + 255) / 256, 256, 0, stream>>>(ei, a_src, a_dst, amax);
  // 4) segment sum of exp
  gat_edge_denom<<<(E_TOT + 255) / 256, 256, 0, stream>>>(ei, a_src, a_dst, amax, denom);
  // 5) weighted scatter aggregate
  {
    const long long tot = (long long)E_TOT * OUT_CH;
    const int blocks = (int)((tot + 255) / 256);
    gat_edge_agg<<<blocks, 256, 0, stream>>>(ei, a_src, a_dst, amax, denom, h, acc);
  }
  // 6) bias + ELU
  {
    const int n = N_NODES * OUT_CH;
    gat_finalize<<<(n + 255) / 256, 256, 0, stream>>>(acc, bias, out);
  }
}